// JointCategorical_3848290697225
// MI455X (gfx1250) — compile-verified
//
#include <hip/hip_runtime.h>

// JointCategorical lookup: out[i] = probs[X[i,0], X[i,1], X[i,2], X[i,3]]
//   probs: 50^4 f32 = 25 MB  -> L2-resident on MI455X (192 MB global L2)
//   X:     8M x 4 int64 = 256 MB, streamed exactly once
//   out:   8M f32 = 32 MB, streamed once
//
// Roofline: (256+32+25) MB / 23.3 TB/s ~ 13.4 us; zero matrix FLOPs -> WMMA
// inapplicable. The CDNA5 feature that *does* apply is the async memory path:
// GLOBAL_LOAD_ASYNC_TO_LDS_B128 (ASYNCcnt) streams X into a double-buffered
// LDS FIFO behind the wave, so the only load on the VGPR dependence chain is
// the L2-hit probs gather. NT temporal hints keep the X/out streams from
// evicting the probs table out of L2.

#define BLOCK 256
#define SAMPLES_PER_THREAD 8
#define BUF_BYTES (BLOCK * 32)   // 32 B (4 x i64) per lane per buffer = 8 KB

__global__ void __launch_bounds__(BLOCK)
joint_categorical_gather_async(const float* __restrict__ probs,
                               const unsigned long long* __restrict__ X,
                               float* __restrict__ out,
                               int n)
{
    // Double-buffered staging tile: 2 x 8 KB = 16 KB per block (<< 320 KB/WGP).
    __shared__ unsigned long long xtile[2][BLOCK * 4];

    const unsigned tid    = threadIdx.x;
    const unsigned stride = gridDim.x * blockDim.x;
    const unsigned i0     = blockIdx.x * blockDim.x + tid;

    // LDS byte offset of the tile inside this workgroup's LDS allocation:
    // the low 32 bits of a generic pointer to an LDS object are the
    // LDS_BASE-relative offset, which is exactly what the async-load VDST
    // operand expects.
    const unsigned lds_base =
        (unsigned)(unsigned long long)(uintptr_t)(&xtile[0][0]);
    const unsigned my_slot = lds_base + tid * 32u;

    // Issue one 32-byte async fill (two b128 ops; offset applies to both the
    // global and the LDS address, per the ISA's async-load pseudocode).
    auto issue_fill = [&](unsigned i, unsigned buf) {
        unsigned long long g = (unsigned long long)(uintptr_t)(X + 4ull * i);
        unsigned           l = my_slot + buf * (unsigned)BUF_BYTES;
        asm volatile(
            "global_load_async_to_lds_b128 %0, %1, off th:TH_LOAD_NT\n\t"
            "global_load_async_to_lds_b128 %0, %1, off offset:16 th:TH_LOAD_NT"
            :: "v"(l), "v"(g) : "memory");
    };

    unsigned buf = 0;
    unsigned i   = i0;
    if (i < (unsigned)n) issue_fill(i, 0);        // prologue: fill buffer 0

    for (; i < (unsigned)n; i += stride) {
        const unsigned inext = i + stride;

        if (inext < (unsigned)n) {
            // Stage next sample into the other buffer while we consume this one.
            issue_fill(inext, buf ^ 1u);
            // Warm GL2 two iterations ahead for the async engine (speculative
            // prefetch: out-of-range addresses are silently dropped).
            __builtin_prefetch((const void*)(X + 4ull * i + 8ull * stride), 0, 0);
            // ASYNCcnt completes in order: <=2 outstanding means the two ops
            // of the *current* buffer have landed in LDS.
            asm volatile("s_wait_asynccnt 0x2" ::: "memory");
        } else {
            asm volatile("s_wait_asynccnt 0x0" ::: "memory");
        }

        // Consume: only the low dword of each little-endian int64 matters
        // (values are in [0,50)) -> 4 x ds_load_b32.
        const unsigned* p = (const unsigned*)&xtile[buf][tid * 4u];
        const unsigned x0 = p[0], x1 = p[2], x2 = p[4], x3 = p[6];

        // flat = ((x0*50 + x1)*50 + x2)*50 + x3   (max 6,249,999 -> u32)
        const unsigned flat = ((x0 * 50u + x1) * 50u + x2) * 50u + x3;

        // Regular-temporal gather: 25 MB table stays L2-resident.
        const float pr = probs[flat];

        // Streaming NT store.
        __builtin_nontemporal_store(pr, out + i);

        buf ^= 1u;
    }
}

extern "C" void kernel_launch(void* const* d_in, const int* in_sizes, int n_in,
                              void* d_out, int out_size, void* d_ws, size_t ws_size,
                              hipStream_t stream)
{
    (void)in_sizes; (void)n_in; (void)d_ws; (void)ws_size;

    const float*              probs = (const float*)d_in[0];
    const unsigned long long* X     = (const unsigned long long*)d_in[1];
    float*                    out   = (float*)d_out;

    const int n = out_size;                      // 8,000,000 samples
    int grid = (n + BLOCK * SAMPLES_PER_THREAD - 1) / (BLOCK * SAMPLES_PER_THREAD);
    if (grid < 1) grid = 1;                      // ~1M threads x 8-deep pipeline

    hipLaunchKernelGGL(joint_categorical_gather_async, dim3(grid), dim3(BLOCK),
                       0, stream, probs, X, out, n);
}